// GRUMemory_62775241999069
// MI455X (gfx1250) — compile-verified
//
#include <hip/hip_runtime.h>
#include <math.h>

#define N_NODES 100000
#define DIM 128
#define N_EDGES 200000
#define KW_IH 512   // message width: 2*DIM_MEM + DIM_EDGE + DIM_TIME
#define KW_HH 128

typedef __attribute__((ext_vector_type(2))) float v2f;
typedef __attribute__((ext_vector_type(8))) float v8f;

// D = A(16x4) * B(4x16) + C(16x16), all f32 — CDNA5 V_WMMA_F32_16X16X4_F32
#define WMMA_F32(A, B, C) \
  __builtin_amdgcn_wmma_f32_16x16x4_f32(false, (A), false, (B), (short)0, (C), false, false)

// LDS row strides padded by 4 floats: row m starts at bank 4m (mod 64), so the
// 32-lane b64 A-fragment read hits all 64 banks exactly once.
#define SA_STRIDE 516
#define SH_STRIDE 132
#define LDS_FLOATS (32 * SA_STRIDE + 32 * SH_STRIDE)

__global__ void initLP(int* lp, int n) {
  int i = blockIdx.x * blockDim.x + threadIdx.x;
  if (i < n) lp[i] = -1;
}

__global__ void scatterLP(const int* __restrict__ src, const int* __restrict__ dst,
                          int* lps, int* lpd, int E) {
  int e = blockIdx.x * blockDim.x + threadIdx.x;
  if (e < E) {
    atomicMax(&lps[src[e]], e);   // 'last' reducer == max edge position
    atomicMax(&lpd[dst[e]], e);
  }
}

__global__ __launch_bounds__(128) void gruPass(
    const float* __restrict__ memOrig, const float* __restrict__ luOrig,
    const float* __restrict__ W_ih, const float* __restrict__ W_hh,
    const float* __restrict__ b_ih, const float* __restrict__ b_hh,
    const float* __restrict__ eTimes, const float* __restrict__ eFeat,
    const int* __restrict__ otherNids, const int* __restrict__ lastPos,
    const float* __restrict__ hIn, float* __restrict__ outMem,
    float* __restrict__ outLu)
{
  extern __shared__ float lds[];
  float* sA = lds;                  // [32][516] message rows (K = 512)
  float* sH = lds + 32 * SA_STRIDE; // [32][132] h rows       (K = 128)
  __shared__ int sE[32];

  const int tid  = threadIdx.x;
  const int base = blockIdx.x * 32;   // 100000 / 32 == 3125 exactly

  if (tid < 32) sE[tid] = lastPos[base + tid];
  __syncthreads();

  // ---------------- stage tile into LDS (4 threads per row) ----------------
  {
    const int r = tid >> 2;
    const int q = tid & 3;
    const int e = sE[r];
    const bool act = (e >= 0);
    int other = 0;
    float dt = 0.f;
    if (act) {
      other = otherNids[e];
      dt = eTimes[e] - luOrig[base + r];   // encoder uses ORIGINAL last_update
    }
    const float4* selfP  = (const float4*)(memOrig + (size_t)(base + r) * DIM);
    const float4* otherP = (const float4*)(memOrig + (size_t)other * DIM);
    const float4* featP  = (const float4*)(eFeat + (size_t)(act ? e : 0) * DIM);
    const float4* hP     = (const float4*)(hIn + (size_t)(base + r) * DIM);
    float* rowA = sA + r * SA_STRIDE;
    float* rowH = sH + r * SH_STRIDE;
    const float4 z4 = make_float4(0.f, 0.f, 0.f, 0.f);
    #pragma unroll
    for (int i = q; i < 32; i += 4) {
      *(float4*)(rowA + i * 4)       = act ? selfP[i]  : z4;  // mem[self]
      *(float4*)(rowA + 128 + i * 4) = act ? otherP[i] : z4;  // mem[other]
      *(float4*)(rowA + 256 + i * 4) = act ? featP[i]  : z4;  // edge features
      float4 enc = z4;
      if (act) {
        #pragma unroll
        for (int jj = 0; jj < 4; ++jj) {
          const int j = i * 4 + jj;
          const float wj = expf((float)j * (-9.0f / 127.0f) * 2.3025850929940457f);
          ((float*)&enc)[jj] = cosf(dt * wj);
        }
      }
      *(float4*)(rowA + 384 + i * 4) = enc;                   // time encoding
      *(float4*)(rowH + i * 4)       = hP[i];                 // GRU hidden in
    }
  }
  if (tid < 32 && sE[tid] >= 0) outLu[base + tid] = eTimes[sE[tid]];
  __syncthreads();

  // ---------------- WMMA GEMM: gi (K=512) and gh (K=128) -------------------
  const int wv   = tid >> 5;          // wave 0..3
  const int lane = tid & 31;
  const int lm   = lane & 15;
  const int kh2  = (lane >> 4) * 2;   // K sub-offset 0 or 2 (A/B frag layout)

  for (int ct = wv; ct < 8; ct += 4) {        // output column tile (16 cols)
    // biases land in the C accumulator; r/z gates fold b_ih + b_hh
    const float bR  = b_ih[ct * 16 + lm]       + b_hh[ct * 16 + lm];
    const float bZ  = b_ih[128 + ct * 16 + lm] + b_hh[128 + ct * 16 + lm];
    const float bIN = b_ih[256 + ct * 16 + lm];
    const float bHN = b_hh[256 + ct * 16 + lm];
    v8f aR0, aR1, aZ0, aZ1, aN0, aN1, aH0, aH1;
    #pragma unroll
    for (int v = 0; v < 8; ++v) {
      aR0[v] = bR;  aR1[v] = bR;  aZ0[v] = bZ;  aZ1[v] = bZ;
      aN0[v] = bIN; aN1[v] = bIN; aH0[v] = bHN; aH1[v] = bHN;
    }
    // B = W^T : lane reads W row (gate_base + ct*16 + lane%16), consecutive k
    const float* Wr = W_ih + (size_t)(ct * 16 + lm) * KW_IH;
    const float* Wz = W_ih + (size_t)(128 + ct * 16 + lm) * KW_IH;
    const float* Wn = W_ih + (size_t)(256 + ct * 16 + lm) * KW_IH;
    const float* A0 = sA + lm * SA_STRIDE;
    const float* A1 = sA + (16 + lm) * SA_STRIDE;
    for (int kb = 0; kb < KW_IH; kb += 4) {
      const int kc = kb + kh2;
      const v2f a0 = *(const v2f*)(A0 + kc);
      const v2f a1 = *(const v2f*)(A1 + kc);
      const v2f br = *(const v2f*)(Wr + kc);
      const v2f bz = *(const v2f*)(Wz + kc);
      const v2f bn = *(const v2f*)(Wn + kc);
      aR0 = WMMA_F32(a0, br, aR0);
      aR1 = WMMA_F32(a1, br, aR1);
      aZ0 = WMMA_F32(a0, bz, aZ0);
      aZ1 = WMMA_F32(a1, bz, aZ1);
      aN0 = WMMA_F32(a0, bn, aN0);   // i_n kept separate
      aN1 = WMMA_F32(a1, bn, aN1);
    }
    const float* Vr = W_hh + (size_t)(ct * 16 + lm) * KW_HH;
    const float* Vz = W_hh + (size_t)(128 + ct * 16 + lm) * KW_HH;
    const float* Vn = W_hh + (size_t)(256 + ct * 16 + lm) * KW_HH;
    const float* H0 = sH + lm * SH_STRIDE;
    const float* H1 = sH + (16 + lm) * SH_STRIDE;
    for (int kb = 0; kb < KW_HH; kb += 4) {
      const int kc = kb + kh2;
      const v2f a0 = *(const v2f*)(H0 + kc);
      const v2f a1 = *(const v2f*)(H1 + kc);
      const v2f br = *(const v2f*)(Vr + kc);
      const v2f bz = *(const v2f*)(Vz + kc);
      const v2f bn = *(const v2f*)(Vn + kc);
      aR0 = WMMA_F32(a0, br, aR0);   // r,z gates accumulate gi + gh
      aR1 = WMMA_F32(a1, br, aR1);
      aZ0 = WMMA_F32(a0, bz, aZ0);
      aZ1 = WMMA_F32(a1, bz, aZ1);
      aH0 = WMMA_F32(a0, bn, aH0);   // h_n kept separate
      aH1 = WMMA_F32(a1, bn, aH1);
    }
    // ------------- elementwise GRU + masked store (C/D layout) -------------
    const int cn = lm;                 // n = lane % 16
    const int mh = (lane >> 4) * 8;    // m = v + 8*(lane>=16)
    #pragma unroll
    for (int mt = 0; mt < 2; ++mt) {
      const v8f vR = mt ? aR1 : aR0;
      const v8f vZ = mt ? aZ1 : aZ0;
      const v8f vN = mt ? aN1 : aN0;
      const v8f vH = mt ? aH1 : aH0;
      #pragma unroll
      for (int v = 0; v < 8; ++v) {
        const int m = mt * 16 + mh + v;
        const float r  = 1.0f / (1.0f + expf(-vR[v]));
        const float z  = 1.0f / (1.0f + expf(-vZ[v]));
        const float nn = tanhf(vN[v] + r * vH[v]);
        const float hc = sH[m * SH_STRIDE + ct * 16 + cn];
        const float hnew = (1.0f - z) * nn + z * hc;
        if (sE[m] >= 0)
          outMem[(size_t)(base + m) * DIM + ct * 16 + cn] = hnew;
      }
    }
  }
}

extern "C" void kernel_launch(void* const* d_in, const int* in_sizes, int n_in,
                              void* d_out, int out_size, void* d_ws, size_t ws_size,
                              hipStream_t stream) {
  const float* memory      = (const float*)d_in[0];
  const float* last_update = (const float*)d_in[1];
  const float* W_ih        = (const float*)d_in[2];
  const float* W_hh        = (const float*)d_in[3];
  const float* b_ih        = (const float*)d_in[4];
  const float* b_hh        = (const float*)d_in[5];
  const float* eTimes      = (const float*)d_in[6];
  const float* eFeat       = (const float*)d_in[7];
  const int*   src         = (const int*)d_in[8];
  const int*   dst         = (const int*)d_in[9];

  float* outMem = (float*)d_out;
  float* outLu  = outMem + (size_t)N_NODES * DIM;
  int* lps = (int*)d_ws;
  int* lpd = lps + N_NODES;

  const size_t ldsBytes = (size_t)LDS_FLOATS * sizeof(float);
  (void)hipFuncSetAttribute((const void*)gruPass,
                            hipFuncAttributeMaxDynamicSharedMemorySize,
                            (int)ldsBytes);

  initLP<<<(2 * N_NODES + 255) / 256, 256, 0, stream>>>(lps, 2 * N_NODES);
  scatterLP<<<(N_EDGES + 255) / 256, 256, 0, stream>>>(src, dst, lps, lpd, N_EDGES);

  // seed outputs with original memory / last_update (inactive nodes keep them)
  hipMemcpyAsync(outMem, memory, (size_t)N_NODES * DIM * sizeof(float),
                 hipMemcpyDeviceToDevice, stream);
  hipMemcpyAsync(outLu, last_update, (size_t)N_NODES * sizeof(float),
                 hipMemcpyDeviceToDevice, stream);

  // pass 1 (src): h = original memory; messages from original state
  gruPass<<<N_NODES / 32, 128, ldsBytes, stream>>>(
      memory, last_update, W_ih, W_hh, b_ih, b_hh, eTimes, eFeat,
      dst, lps, memory, outMem, outLu);
  // pass 2 (dst): h = post-src memory (outMem); messages STILL from originals
  gruPass<<<N_NODES / 32, 128, ldsBytes, stream>>>(
      memory, last_update, W_ih, W_hh, b_ih, b_hh, eTimes, eFeat,
      src, lpd, outMem, outMem, outLu);
}